// Sinkhorn_55594056680009
// MI455X (gfx1250) — compile-verified
//
#include <hip/hip_runtime.h>
#include <math.h>

// Sinkhorn (log-domain), b=64, n=1024, tau=1, 10 iters.
// log_s stays rank-structured: log_s = s/tau - r[b,i] - c[b,j].
// 5x row passes (WMMA f32 16x16x4 row-sum reduction), 5x col passes
// (online LSE, coalesced), 1 exp epilogue. Bandwidth-bound: ~512MB HBM
// with L2 reuse of the constant 256MB input.

typedef float v2f __attribute__((ext_vector_type(2)));
typedef float v4f __attribute__((ext_vector_type(4)));
typedef float v8f __attribute__((ext_vector_type(8)));

#define BATCH 64
#define NDIM  1024
#define INV_TAU 1.0f

__global__ void sk_init_c(float* __restrict__ c) {
    int idx = blockIdx.x * 256 + threadIdx.x;
    c[idx] = 0.0f;
}

// Row pass: r[b,i] = logsumexp_j( s[b,i,j]*INV_TAU - c[b,j] )
// One wave (32 lanes) per 16 rows. A-matrix layout for V_WMMA_F32_16X16X4_F32:
//   lanes 0-15 : M=lane, VGPR0/1 = K=0,1
//   lanes 16-31: M=lane-16, VGPR0/1 = K=2,3
// B = all-ones 4x16 (layout irrelevant), C/D = f32 16x16 accumulator.
__global__ void __launch_bounds__(32)
sk_row_pass(const float* __restrict__ s, const float* __restrict__ c,
            float* __restrict__ r) {
    const int lane = threadIdx.x;                 // 0..31
    const int b    = blockIdx.x >> 6;             // 64 row-groups per batch
    const int rg   = blockIdx.x & 63;
    const int row  = (rg << 4) | (lane & 15);
    const int koff = (lane & 16) ? 2 : 0;

    const float* __restrict__ srow = s + ((size_t)b << 20) + ((size_t)row << 10);
    const float* __restrict__ cb   = c + (b << 10);

    // Phase 1: row max of x = s*invtau - c[j] (lanes L and L+16 jointly cover row L)
    float m = -INFINITY;
    for (int k = 0; k < NDIM; k += 4) {
        v2f sv = *(const v2f*)(srow + k + koff);
        v2f cv = *(const v2f*)(cb + k + koff);
        float x0 = sv.x * INV_TAU - cv.x;
        float x1 = sv.y * INV_TAU - cv.y;
        m = fmaxf(m, fmaxf(x0, x1));
    }
    m = fmaxf(m, __shfl_xor(m, 16, 32));          // combine K-halves of each row

    // Phase 2: sum of exp(x - m) via f32 WMMA against a ones matrix.
    v8f acc = {0.f, 0.f, 0.f, 0.f, 0.f, 0.f, 0.f, 0.f};
    const v2f ones = {1.0f, 1.0f};
    for (int k = 0; k < NDIM; k += 4) {
        v2f sv = *(const v2f*)(srow + k + koff);
        v2f cv = *(const v2f*)(cb + k + koff);
        v2f a;
        a.x = __expf(sv.x * INV_TAU - cv.x - m);
        a.y = __expf(sv.y * INV_TAU - cv.y - m);
        // D = A x ones + C : row sums replicated across N, f32 accumulate.
        acc = __builtin_amdgcn_wmma_f32_16x16x4_f32(
            /*neg_a=*/false, a, /*neg_b=*/false, ones,
            /*c_mod=*/(short)0, acc, /*reuse_a=*/false, /*reuse_b=*/false);
    }

    // D layout: VGPR t, lanes 0-15 -> M=t ; lanes 16-31 -> M=8+t.
    __shared__ float ssum[16];
    if (lane == 0) {
        ssum[0] = acc[0]; ssum[1] = acc[1]; ssum[2] = acc[2]; ssum[3] = acc[3];
        ssum[4] = acc[4]; ssum[5] = acc[5]; ssum[6] = acc[6]; ssum[7] = acc[7];
    }
    if (lane == 16) {
        ssum[8]  = acc[0]; ssum[9]  = acc[1]; ssum[10] = acc[2]; ssum[11] = acc[3];
        ssum[12] = acc[4]; ssum[13] = acc[5]; ssum[14] = acc[6]; ssum[15] = acc[7];
    }
    __syncthreads();
    if (lane < 16) {   // lane L already holds row L's max m
        r[(b << 10) + (rg << 4) + lane] = m + __logf(ssum[lane]);
    }
}

// Col pass: c[b,j] = logsumexp_i( s[b,i,j]*INV_TAU - r[b,i] )
// Block of 1024 threads handles (batch b, 256-column tile); i-range split into
// 4 chunks of 256 for shorter dependency chains; partials merged via LDS.
__global__ void __launch_bounds__(1024)
sk_col_pass(const float* __restrict__ s, const float* __restrict__ r,
            float* __restrict__ c) {
    __shared__ float rs[NDIM];
    __shared__ float pm[4][256];
    __shared__ float ps[4][256];

    const int b   = blockIdx.x >> 2;
    const int j0  = (blockIdx.x & 3) << 8;
    const int tid = threadIdx.x;
    const int t   = tid & 255;        // column within tile
    const int ci  = tid >> 8;         // i-chunk 0..3

    rs[tid] = r[(b << 10) + tid];
    __syncthreads();

    const float* __restrict__ sb = s + ((size_t)b << 20) + j0 + t;
    float m = -INFINITY, sum = 0.0f;
    const int i0 = ci << 8;
    for (int i = i0; i < i0 + 256; ++i) {
        float x  = sb[(size_t)i << 10] * INV_TAU - rs[i];
        float mn = fmaxf(m, x);
        sum = sum * __expf(m - mn) + __expf(x - mn);   // online LSE
        m = mn;
    }
    pm[ci][t] = m;
    ps[ci][t] = sum;
    __syncthreads();

    if (tid < 256) {
        float m0 = pm[0][t], m1 = pm[1][t], m2 = pm[2][t], m3 = pm[3][t];
        float mm = fmaxf(fmaxf(m0, m1), fmaxf(m2, m3));
        float ss = ps[0][t] * __expf(m0 - mm) + ps[1][t] * __expf(m1 - mm)
                 + ps[2][t] * __expf(m2 - mm) + ps[3][t] * __expf(m3 - mm);
        c[(b << 10) + j0 + t] = mm + __logf(ss);
    }
}

// Epilogue: out = exp(s*INV_TAU - r[b,i] - c[b,j]), float4-vectorized.
__global__ void __launch_bounds__(256)
sk_final(const float* __restrict__ s, const float* __restrict__ r,
         const float* __restrict__ cc, float* __restrict__ out) {
    size_t q = (size_t)blockIdx.x * 256 + threadIdx.x;   // quad index
    size_t e = q << 2;                                   // element index
    int b   = (int)(e >> 20);
    int rem = (int)(e & ((1u << 20) - 1));
    int i   = rem >> 10;
    int j   = rem & 1023;

    v4f sv  = *(const v4f*)(s + e);
    v4f cv  = *(const v4f*)(cc + (b << 10) + j);
    float rv = r[(b << 10) + i];

    v4f o;
    o.x = __expf(sv.x * INV_TAU - rv - cv.x);
    o.y = __expf(sv.y * INV_TAU - rv - cv.y);
    o.z = __expf(sv.z * INV_TAU - rv - cv.z);
    o.w = __expf(sv.w * INV_TAU - rv - cv.w);
    *(v4f*)(out + e) = o;
}

extern "C" void kernel_launch(void* const* d_in, const int* in_sizes, int n_in,
                              void* d_out, int out_size, void* d_ws, size_t ws_size,
                              hipStream_t stream) {
    const float* s = (const float*)d_in[0];
    float* out = (float*)d_out;

    // Workspace: r then c, each BATCH*NDIM floats (256 KB each).
    float* r = (float*)d_ws;
    float* c = r + (size_t)BATCH * NDIM;

    sk_init_c<<<(BATCH * NDIM) / 256, 256, 0, stream>>>(c);

    // 10 reference iterations = 5 x (row norm, col norm)
    for (int it = 0; it < 5; ++it) {
        sk_row_pass<<<BATCH * (NDIM / 16), 32, 0, stream>>>(s, c, r);
        sk_col_pass<<<BATCH * (NDIM / 256), 1024, 0, stream>>>(s, r, c);
    }

    sk_final<<<BATCH * NDIM * (NDIM / 4) / 256, 256, 0, stream>>>(s, r, c, out);
}